// MaxUnpooling2D_52673478918313
// MI455X (gfx1250) — compile-verified
//
#include <hip/hip_runtime.h>

// Max-unpooling scatter-add, MI455X (gfx1250, wave32).
// B=16, H=128, W=128, C=64, OH=OW=256.
// Input linear index  i = ((b*H + h)*W + w)*C + c  -> per-batch 2^20 elems.
// Output linear index o = ((b*OH + y)*OW + x)*C + c -> per-batch 2^22 elems.
// argmax decode (all powers of two): y = (a >> 14) & 255, x = (a >> 6) & 255.
//
// NOTE: unpool_scatter is defined FIRST so the disasm snippet shows it:
// expect 2x global_load_b128 th:NT + 4x non-returning global_atomic_add_f32.

typedef float v4f __attribute__((ext_vector_type(4)));
typedef int   v4i __attribute__((ext_vector_type(4)));

// ---------------------------------------------------------------------------
// Scatter-add. One thread handles 4 consecutive channels:
//   - 1x global_load_b128 (updates, NT) + 1x global_load_b128 (argmax, NT)
//   - 4x non-returning global_atomic_add_f32 to decoded addresses
// NT hints keep the streamed 128 MB of inputs from evicting the (mostly
// L2-resident) 256 MB output between the zero-fill and atomic phases.
// ---------------------------------------------------------------------------
__global__ __launch_bounds__(256) void unpool_scatter(
    const v4f* __restrict__ updates4,
    const v4i* __restrict__ argmax4,
    float* __restrict__ out,
    int n4) {
    int t = blockIdx.x * blockDim.x + threadIdx.x;
    if (t >= n4) return;

    v4f u = __builtin_nontemporal_load(updates4 + t);  // th:NT
    v4i a = __builtin_nontemporal_load(argmax4  + t);  // th:NT

    unsigned i0   = (unsigned)t << 2;      // base element index (multiple of 4)
    unsigned b    = i0 >> 20;              // batch  (H*W*C = 2^20 per batch)
    unsigned c    = i0 & 63u;              // channel base (C = 64; 4-aligned)
    unsigned base = (b << 22) + c;         // OH*OW*C = 2^22 per output batch

#pragma unroll
    for (int j = 0; j < 4; ++j) {
        unsigned av  = (unsigned)a[j];
        unsigned y   = (av >> 14) & 255u;  // a / (OW*C) % OH
        unsigned x   = (av >> 6)  & 255u;  // a / C % OW
        unsigned off = base + (y << 14) + (x << 6) + (unsigned)j;
        // Native non-returning f32 atomic add (sums duplicate targets).
        (void)unsafeAtomicAdd(out + off, u[j]);
    }
}

// ---------------------------------------------------------------------------
// Zero the 256 MB output with 128-bit stores, regular temporal hint
// (leave zeroed lines resident in L2 so the atomic phase hits in L2).
// ---------------------------------------------------------------------------
__global__ __launch_bounds__(256) void unpool_zero_fill(v4f* __restrict__ out,
                                                        int n4) {
    int i = blockIdx.x * blockDim.x + threadIdx.x;
    if (i < n4) {
        out[i] = (v4f)(0.0f);   // global_store_b128
    }
}

// ---------------------------------------------------------------------------
extern "C" void kernel_launch(void* const* d_in, const int* in_sizes, int n_in,
                              void* d_out, int out_size, void* d_ws, size_t ws_size,
                              hipStream_t stream) {
    const v4f* updates4 = (const v4f*)d_in[0];  // float32 [16,128,128,64]
    const v4i* argmax4  = (const v4i*)d_in[1];  // int32   [16,128,128,64]
    float*     out      = (float*)d_out;        // float32 [16,256,256,64]

    const int n_in_elems = in_sizes[0];         // 16,777,216
    const int n4_in      = n_in_elems >> 2;     //  4,194,304 scatter threads
    const int n4_out     = out_size >> 2;       // 16,777,216 float4 zero-fills

    const int BLOCK = 256;                      // 8 wave32 waves / block

    unpool_zero_fill<<<(n4_out + BLOCK - 1) / BLOCK, BLOCK, 0, stream>>>(
        (v4f*)d_out, n4_out);

    unpool_scatter<<<(n4_in + BLOCK - 1) / BLOCK, BLOCK, 0, stream>>>(
        updates4, argmax4, out, n4_in);
}